// GCNEncoder_81707457839461
// MI455X (gfx1250) — compile-verified
//
#include <hip/hip_runtime.h>
#include <hip/hip_bf16.h>

typedef __attribute__((ext_vector_type(2))) float v2f;
typedef __attribute__((ext_vector_type(8))) float v8f;

#define KCHUNK 64   // K-depth staged in LDS per step
#define KPAD   66   // LDS row pitch (dwords): even (b64 align) + bank spread

// ---------------------------------------------------------------- utilities

__global__ void zero_f32_kernel(float* __restrict__ p, int n) {
  int i = blockIdx.x * blockDim.x + threadIdx.x;
  if (i < n) p[i] = 0.0f;
}

__global__ void deg_count_kernel(const int* __restrict__ dst, float* __restrict__ deg, int E) {
  int e = blockIdx.x * blockDim.x + threadIdx.x;
  if (e < E) atomicAdd(&deg[dst[e]], 1.0f);
}

__global__ void deg_to_dinv_kernel(float* __restrict__ d, int n) {
  int i = blockIdx.x * blockDim.x + threadIdx.x;
  if (i < n) d[i] = rsqrtf(d[i] + 1.0f);
}

// feat = [emb_a[cat_a] | emb_b[cat_b] | x[:,2:10]]  -> (N, IN) with IN = 2C+8
__global__ void build_feat_kernel(const float* __restrict__ x,
                                  const float* __restrict__ ea,
                                  const float* __restrict__ eb,
                                  float* __restrict__ feat,
                                  int N, int C, int IN) {
  int t = blockIdx.x * blockDim.x + threadIdx.x;
  if (t >= N * IN) return;
  int i = t / IN;
  int j = t - i * IN;
  float v;
  if (j < C) {
    int ia = (int)x[i * 10 + 0];
    v = ea[ia * C + j];
  } else if (j < 2 * C) {
    int ib = (int)x[i * 10 + 1];
    v = eb[ib * C + (j - C)];
  } else {
    v = x[i * 10 + 2 + (j - 2 * C)];
  }
  feat[t] = v;
}

// -------------------------------------------------------------- WMMA GEMM
// C[M,NCOLS] = A[M,K](lda) * B[K,NCOLS], fp32 via V_WMMA_F32_16X16X4_F32.
// NCOLS is a template constant: stores become base + immediate offsets and
// the staging loop uses constant shifts. One wave per 16x16 tile; B staged
// transposed in LDS (sB[col][k], pitch KPAD) so each lane's K-pair is a
// single ds_load_b64; A fragments are single global_load_b64. Full KCHUNK
// slices are fully unrolled (16 WMMAs, clause-able loads). No early returns:
// every wave reaches every __syncthreads; EXEC is all-1s at each WMMA.
template <int NCOLS>
__global__ void wmma_gemm_f32_lds_kernel(const float* __restrict__ A,
                                         const float* __restrict__ B,
                                         float* __restrict__ Cout,
                                         int M, int K, int lda) {
  constexpr int NSHIFT = (NCOLS == 128) ? 7 : 6;
  constexpr int TILESN = NCOLS >> 4;
  __shared__ float sB[NCOLS * KPAD];

  const int lane   = threadIdx.x & 31;
  const int wave   = threadIdx.x >> 5;
  const int wavesPerBlock = blockDim.x >> 5;
  const int tilesM = (M + 15) >> 4;
  const int strips = wavesPerBlock / TILESN;        // M-strips per block

  const int tile_n = wave & (TILESN - 1);
  const int strip  = blockIdx.x * strips + (wave >> (NSHIFT - 5));
  const bool active = strip < tilesM;

  const int tm = (active ? strip : 0) << 4;
  const int tn = tile_n << 4;
  const int l16    = lane & 15;
  const int half16 = lane >> 4;                     // K pair {0,1} vs {2,3}

  int arow = tm + l16;
  if (arow >= M) arow = M - 1;
  const int bcol = tn + l16;

  const float* __restrict__ Abase = A + (size_t)arow * lda + (half16 << 1);
  const float* __restrict__ Bbase = sB + bcol * KPAD + (half16 << 1);

  v8f acc = {};
  for (int k0 = 0; k0 < K; k0 += KCHUNK) {
    const int kc = (K - k0) < KCHUNK ? (K - k0) : KCHUNK;
    __syncthreads();
    // stage B[k0:k0+kc, :] transposed into LDS: float4 global loads,
    // 4 scalar LDS stores each (one-time transpose conflicts are fine).
    for (int idx = threadIdx.x; idx < ((kc << NSHIFT) >> 2); idx += blockDim.x) {
      const int kk = idx >> (NSHIFT - 2);
      const int c4 = (idx & ((NCOLS >> 2) - 1)) << 2;
      const float4 w = *reinterpret_cast<const float4*>(
          B + (size_t)(k0 + kk) * NCOLS + c4);
      sB[(c4 + 0) * KPAD + kk] = w.x;
      sB[(c4 + 1) * KPAD + kk] = w.y;
      sB[(c4 + 2) * KPAD + kk] = w.z;
      sB[(c4 + 3) * KPAD + kk] = w.w;
    }
    __syncthreads();

    const float* __restrict__ Ap = Abase + k0;
    if (kc == KCHUNK) {
#pragma unroll
      for (int k = 0; k < KCHUNK; k += 4) {
        const v2f a = *(const v2f*)(Ap + k);        // global_load_b64
        const v2f b = *(const v2f*)(Bbase + k);     // ds_load_b64
        acc = __builtin_amdgcn_wmma_f32_16x16x4_f32(false, a, false, b,
                                                    (short)0, acc, false, false);
      }
    } else {
      for (int k = 0; k < kc; k += 4) {
        const v2f a = *(const v2f*)(Ap + k);
        const v2f b = *(const v2f*)(Bbase + k);
        acc = __builtin_amdgcn_wmma_f32_16x16x4_f32(false, a, false, b,
                                                    (short)0, acc, false, false);
      }
    }
  }

  if (active) {
    // C/D layout: lane -> N=bcol; VGPR v -> M = tm + v + 8*(lane>=16)
    float* __restrict__ Cbase =
        Cout + (size_t)(tm + (half16 << 3)) * NCOLS + bcol;
    if (tm + 16 <= M) {
#pragma unroll
      for (int v = 0; v < 8; ++v) Cbase[v * NCOLS] = acc[v];   // imm offsets
    } else {
#pragma unroll
      for (int v = 0; v < 8; ++v)
        if (tm + v + (half16 << 3) < M) Cbase[v * NCOLS] = acc[v];
    }
  }
}

// ---------------------------------------------------------- GCN aggregation

// agg = h * dinv^2   (self term)
__global__ void init_agg_kernel(const float* __restrict__ h,
                                const float* __restrict__ dinv,
                                float* __restrict__ agg, int N, int Ch) {
  int t = blockIdx.x * blockDim.x + threadIdx.x;
  if (t >= N * Ch) return;
  int i = t / Ch;
  float di = dinv[i];
  agg[t] = h[t] * di * di;
}

// agg[dst] += h[src] * dinv[src] * dinv[dst]   (float4 gather, f32 atomics)
__global__ void edge_scatter_kernel(const int* __restrict__ src,
                                    const int* __restrict__ dst,
                                    const float* __restrict__ dinv,
                                    const float* __restrict__ h,
                                    float* __restrict__ agg,
                                    int E, int Ch, int gshift) {
  int tid = blockIdx.x * blockDim.x + threadIdx.x;
  int e = tid >> gshift;
  if (e >= E) return;
  int c = (tid & ((1 << gshift) - 1)) << 2;
  int s = src[e];
  int d = dst[e];
  float norm = dinv[s] * dinv[d];
  const float4 hv = *reinterpret_cast<const float4*>(h + (size_t)s * Ch + c);
  float* out = agg + (size_t)d * Ch + c;
  atomicAdd(out + 0, hv.x * norm);
  atomicAdd(out + 1, hv.y * norm);
  atomicAdd(out + 2, hv.z * norm);
  atomicAdd(out + 3, hv.w * norm);
}

// out = relu(agg + bias)
__global__ void bias_relu_kernel(const float* __restrict__ agg,
                                 const float* __restrict__ bias,
                                 float* __restrict__ out, int N, int Ch) {
  int t = blockIdx.x * blockDim.x + threadIdx.x;
  if (t >= N * Ch) return;
  int c = t & (Ch - 1);      // Ch is a power of two (128 / 64)
  float v = agg[t] + bias[c];
  out[t] = v > 0.0f ? v : 0.0f;
}

// ------------------------------------------------------------------ driver

extern "C" void kernel_launch(void* const* d_in, const int* in_sizes, int n_in,
                              void* d_out, int out_size, void* d_ws, size_t ws_size,
                              hipStream_t stream) {
  const float* x  = (const float*)d_in[0];
  const int*   ei = (const int*)d_in[1];
  const float* ea = (const float*)d_in[2];
  const float* eb = (const float*)d_in[3];
  const float* W1 = (const float*)d_in[4];
  const float* b1 = (const float*)d_in[5];
  const float* W2 = (const float*)d_in[6];
  const float* b2 = (const float*)d_in[7];

  const int N  = in_sizes[0] / 10;   // 50000
  const int E  = in_sizes[1] / 2;    // 800000
  const int C  = in_sizes[7];        // 64  (b2)
  const int H  = in_sizes[5];        // 128 (b1)
  const int IN = 2 * C + 8;          // 136

  const int* src = ei;
  const int* dst = ei + E;

  // workspace layout (all fp32)
  float* dinv = (float*)d_ws;
  float* feat = dinv + ((N + 63) & ~63);          // N*IN
  float* h    = feat + (size_t)N * IN;            // N*H   (h1, then h2)
  float* agg  = h    + (size_t)N * H;             // N*H   (agg1, then agg2)
  float* hact = feat;                             // reuse feat buf (IN >= H)

  const int BT = 256;
  auto blocks = [](long long n, int bt) { return (unsigned)((n + bt - 1) / bt); };
  auto ilog2  = [](unsigned v) { return 31 - __builtin_clz(v); };

  // 1) degree -> dinv
  zero_f32_kernel<<<blocks(N, BT), BT, 0, stream>>>(dinv, N);
  deg_count_kernel<<<blocks(E, BT), BT, 0, stream>>>(dst, dinv, E);
  deg_to_dinv_kernel<<<blocks(N, BT), BT, 0, stream>>>(dinv, N);

  // 2) feature assembly
  build_feat_kernel<<<blocks((long long)N * IN, BT), BT, 0, stream>>>(
      x, ea, eb, feat, N, C, IN);

  const int tilesM = (N + 15) / 16;

  // 3) h1 = feat @ W1     (WMMA f32, B staged in LDS; 8 waves = 8 N-tiles)
  {
    const int strips = (BT / 32) / (H / 16);         // 1
    const unsigned grid = (unsigned)((tilesM + strips - 1) / strips);
    wmma_gemm_f32_lds_kernel<128><<<grid, BT, 0, stream>>>(feat, W1, h, N, IN, IN);
  }

  // 4) agg1 = self + edge scatter ; hact = relu(agg1 + b1)
  init_agg_kernel<<<blocks((long long)N * H, BT), BT, 0, stream>>>(h, dinv, agg, N, H);
  {
    const int gshift = ilog2((unsigned)(H / 4));     // 5
    edge_scatter_kernel<<<blocks((long long)E << gshift, BT), BT, 0, stream>>>(
        src, dst, dinv, h, agg, E, H, gshift);
  }
  bias_relu_kernel<<<blocks((long long)N * H, BT), BT, 0, stream>>>(agg, b1, hact, N, H);

  // 5) h2 = hact @ W2     (WMMA f32; 8 waves = 4 N-tiles x 2 M-strips)
  {
    const int strips = (BT / 32) / (C / 16);         // 2
    const unsigned grid = (unsigned)((tilesM + strips - 1) / strips);
    wmma_gemm_f32_lds_kernel<64><<<grid, BT, 0, stream>>>(hact, W2, h, N, H, H);
  }

  // 6) agg2 + relu(agg2 + b2) -> d_out
  init_agg_kernel<<<blocks((long long)N * C, BT), BT, 0, stream>>>(h, dinv, agg, N, C);
  {
    const int gshift = ilog2((unsigned)(C / 4));     // 4
    edge_scatter_kernel<<<blocks((long long)E << gshift, BT), BT, 0, stream>>>(
        src, dst, dinv, h, agg, E, C, gshift);
  }
  bias_relu_kernel<<<blocks((long long)N * C, BT), BT, 0, stream>>>(
      agg, b2, (float*)d_out, N, C);
}